// Attention_58514634441262
// MI455X (gfx1250) — compile-verified
//
#include <hip/hip_runtime.h>

// ---------------------------------------------------------------------------
// GAT-style edge attention: h = leakyrelu(x @ W + b)  -> scatter softmax(idx)
// x: [1, E, 128] f32, W: [128,1] f32, b: [1] f32, idx: [E] i32 in [0, NSEG)
// out: [1, E, 1] f32
//
// HBM-bound: 1 GB stream of x dominates (~45us at 23.3 TB/s). Matvec via
// V_WMMA_F32_16X16X4_F32 (full f32), one 16-edge tile per wave, staged
// through LDS (pad stride 132 -> conflict-free ds_read_b64). W is staged
// into LDS once per wave so the k-loop has zero global traffic beyond x.
// ---------------------------------------------------------------------------

typedef __attribute__((ext_vector_type(2))) float v2f;
typedef __attribute__((ext_vector_type(8))) float v8f;

#define NSEG 50000
#define NEG_SLOPE 0.2f
#define D_DIM 128
#define LDS_STRIDE 132            // 128 + 4 pad, multiple of 4 -> aligned float4
#define WAVE_SLICE (16 * LDS_STRIDE + D_DIM)  // x tile + private W copy
#define WAVES_PER_BLOCK 4         // 4 * 2240 floats = 35,840 B LDS / block

// Monotonic order-preserving float<->uint encoding (for integer atomicMax).
__device__ __forceinline__ unsigned enc_f32(float f) {
    unsigned u = __float_as_uint(f);
    return (u & 0x80000000u) ? ~u : (u | 0x80000000u);
}
__device__ __forceinline__ float dec_f32(unsigned u) {
    unsigned b = (u & 0x80000000u) ? (u & 0x7FFFFFFFu) : ~u;
    return __uint_as_float(b);
}

// ---- kernel 0: init segment max (encoded, 0 == "empty") and segment sum ----
__global__ void k_init(unsigned* __restrict__ mbits, float* __restrict__ s) {
    int i = blockIdx.x * blockDim.x + threadIdx.x;
    if (i < NSEG) { mbits[i] = 0u; s[i] = 0.0f; }
}

// ---- kernel 1: per-edge dot(x,W)+b, LeakyReLU, store h, atomic seg max ----
__global__ __launch_bounds__(WAVES_PER_BLOCK * 32)
void k_dot_max(const float* __restrict__ x, const float* __restrict__ W,
               const float* __restrict__ bias, const int* __restrict__ idx,
               float* __restrict__ h_out, unsigned* __restrict__ mbits,
               long long E, int ntiles) {
    __shared__ float lds[WAVES_PER_BLOCK * WAVE_SLICE];

    const int lane = threadIdx.x & 31;
    const int wave = threadIdx.x >> 5;
    const int tile = blockIdx.x * WAVES_PER_BLOCK + wave;
    if (tile >= ntiles) return;

    const long long base_edge = (long long)tile * 16;
    float* tl = lds + wave * WAVE_SLICE;          // 16 x 128 tile (padded)
    float* wl = tl + 16 * LDS_STRIDE;             // this wave's W copy [128]

    // One-time: stage W (512 B) into this wave's LDS slice. 32 lanes x float4.
    *(float4*)(wl + lane * 4) = *(const float4*)(W + lane * 4);

    // Stage 16x128 tile (one contiguous 8KB block of x) into LDS, coalesced.
    const float* xg = x + base_edge * D_DIM;
    #pragma unroll
    for (int i = 0; i < 16; ++i) {
        int f4  = i * 32 + lane;   // float4 index within tile, 0..511
        int gf  = f4 * 4;          // flat float index, 0..2047
        int row = gf >> 7;         // gf / 128
        int col = gf & 127;        // gf % 128
        float4 v;
        if (base_edge + row < E) v = *(const float4*)(xg + gf);
        else                     v = make_float4(0.f, 0.f, 0.f, 0.f);
        *(float4*)(tl + row * LDS_STRIDE + col) = v;  // stride%4==0 -> aligned
    }
    // No barrier: each wave consumes only its own LDS slice (DScnt ordering).

    // A fragment layout for V_WMMA_F32_16X16X4_F32 (16x4, MxK):
    //   VGPR0: lanes 0-15 K=0 (M=lane), lanes 16-31 K=2 (M=lane-16)
    //   VGPR1: lanes 0-15 K=1,          lanes 16-31 K=3
    // B (4x16, KxN) mirrors this with N=lane; we broadcast W across all N
    // (same-address LDS reads within a lane group broadcast, conflict-free).
    const int row  = lane & 15;
    const int koff = (lane >> 4) << 1;   // 0 for lanes 0-15, 2 for 16-31

    v8f c = {};
    #pragma unroll
    for (int k = 0; k < D_DIM; k += 4) {
        const int kk = k + koff;
        v2f a, bm;
        a.x  = tl[row * LDS_STRIDE + kk];
        a.y  = tl[row * LDS_STRIDE + kk + 1];
        bm.x = wl[kk];
        bm.y = wl[kk + 1];
        // D[M,N] += sum_k A[M,k]*B[k,N]; every column N holds the same dot.
        c = __builtin_amdgcn_wmma_f32_16x16x4_f32(
                /*neg_a=*/false, a, /*neg_b=*/false, bm,
                /*c_mod=*/(short)0, c, /*reuse_a=*/false, /*reuse_b=*/false);
    }

    // C/D layout: VGPR r -> M=r (lanes 0-15) and M=r+8 (lanes 16-31),
    // replicated over N. Lane 0 emits rows 0..7, lane 16 emits rows 8..15.
    if ((lane & 15) == 0) {
        const float bv = bias[0];
        const long long rbase = base_edge + (long long)((lane >> 4) << 3);
        #pragma unroll
        for (int r = 0; r < 8; ++r) {
            long long e = rbase + r;
            if (e < E) {
                float h = c[r] + bv;
                h = (h >= 0.0f) ? h : NEG_SLOPE * h;
                h_out[e] = h;
                atomicMax(&mbits[idx[e]], enc_f32(h));
            }
        }
    }
}

// ---- kernel 2: ev = exp(h - m[idx]) in place; atomicAdd into segment sum ---
__global__ void k_exp_sum(const int* __restrict__ idx,
                          const unsigned* __restrict__ mbits,
                          float* __restrict__ h_inout, float* __restrict__ s,
                          long long E) {
    long long e = (long long)blockIdx.x * blockDim.x + threadIdx.x;
    if (e >= E) return;
    int seg = idx[e];
    float m  = dec_f32(mbits[seg]);
    float ev = expf(h_inout[e] - m);
    h_inout[e] = ev;
    atomicAdd(&s[seg], ev);
}

// ---- kernel 3: normalize ---------------------------------------------------
__global__ void k_norm(const int* __restrict__ idx, const float* __restrict__ s,
                       float* __restrict__ h_inout, long long E) {
    long long e = (long long)blockIdx.x * blockDim.x + threadIdx.x;
    if (e >= E) return;
    h_inout[e] = h_inout[e] / s[idx[e]];
}

extern "C" void kernel_launch(void* const* d_in, const int* in_sizes, int n_in,
                              void* d_out, int out_size, void* d_ws, size_t ws_size,
                              hipStream_t stream) {
    const float* x    = (const float*)d_in[0];
    const float* W    = (const float*)d_in[1];
    const float* bias = (const float*)d_in[2];
    const int*   idx  = (const int*)d_in[3];
    float*       out  = (float*)d_out;

    const long long E = (long long)in_sizes[3];          // idx has E entries
    unsigned* mbits = (unsigned*)d_ws;                   // [NSEG]
    float*    s     = (float*)((unsigned*)d_ws + NSEG);  // [NSEG]

    const int ntiles = (int)((E + 15) / 16);

    k_init<<<(NSEG + 255) / 256, 256, 0, stream>>>(mbits, s);
    k_dot_max<<<(ntiles + WAVES_PER_BLOCK - 1) / WAVES_PER_BLOCK,
                WAVES_PER_BLOCK * 32, 0, stream>>>(x, W, bias, idx, out, mbits,
                                                   E, ntiles);
    k_exp_sum<<<(unsigned)((E + 255) / 256), 256, 0, stream>>>(idx, mbits, out, s, E);
    k_norm<<<(unsigned)((E + 255) / 256), 256, 0, stream>>>(idx, s, out, E);
}